// EntmaxBisect_36687610642643
// MI455X (gfx1250) — compile-verified
//
#include <hip/hip_runtime.h>
#include <stdint.h>

#define D       32000
#define DV      (D / 4)        // 8000 float4 per row (128000 B, 16B aligned)
#define NT      512            // 16 wave32 waves per block
#define NWAVES  (NT / 32)
#define CMAX    8              // per-thread candidate regs -> up to 4096 survivors/row
#define N_ITER  50
// (1/32000)^(alpha-1) with alpha=1.5 -> 1/sqrt(32000)
#define TAU_HI_OFS 0.005590169943749474f

__device__ __forceinline__ float wave_sum(float v) {
#pragma unroll
  for (int o = 16; o > 0; o >>= 1) v += __shfl_xor(v, o, 32);
  return v;
}
__device__ __forceinline__ float wave_max(float v) {
#pragma unroll
  for (int o = 16; o > 0; o >>= 1) v = fmaxf(v, __shfl_xor(v, o, 32));
  return v;
}

__global__ __launch_bounds__(NT)
void entmax15_bisect_kernel(const float* __restrict__ X, float* __restrict__ Out) {
  __shared__ float row[D];                 // 128 KB staged row
  __shared__ float red[2][NWAVES + 1];     // double-banked reduction scratch
  __shared__ int   s_over;

  const int tid  = threadIdx.x;
  const int lane = tid & 31;
  const int wv   = tid >> 5;
  const int64_t r = (int64_t)blockIdx.x;
  const float* __restrict__ src = X + r * (int64_t)D;
  float* __restrict__ dst       = Out + r * (int64_t)D;

  if (tid == 0) s_over = 0;

  // ---- CDNA5 async bulk copy: global -> LDS, tracked by ASYNCcnt ----
  {
    const uint32_t lds0  = (uint32_t)(uintptr_t)&row[0];   // low 32 bits of flat LDS addr = LDS offset
    const uint64_t gbase = (uint64_t)(uintptr_t)src;
    for (int v = tid; v < DV; v += NT) {
      const uint32_t lds = lds0 + (uint32_t)v * 16u;
      const uint32_t off = (uint32_t)v * 16u;
      asm volatile("global_load_async_to_lds_b128 %0, %1, %2"
                   :: "v"(lds), "v"(off), "s"(gbase) : "memory");
    }
    asm volatile("s_wait_asynccnt 0x0" ::: "memory");
  }
  __syncthreads();

  const float4* __restrict__ rv = (const float4*)row;

  // ---- row max (bank 1 of reduction scratch) ----
  float m = -INFINITY;
  for (int v = tid; v < DV; v += NT) {
    const float4 q = rv[v];
    m = fmaxf(m, fmaxf(fmaxf(q.x, q.y), fmaxf(q.z, q.w)));
  }
  m = wave_max(m);
  if (lane == 0) red[1][wv] = m;
  __syncthreads();
  if (wv == 0) {
    float t = (lane < NWAVES) ? red[1][lane] : -INFINITY;
    t = wave_max(t);
    if (lane == 0) red[1][NWAVES] = t;
  }
  __syncthreads();
  const float max_s  = 0.5f * red[1][NWAVES];     // max of Xs = 0.5*X
  const float tau_lo0 = max_s - 1.0f;

  // ---- candidate compaction into registers: only Xs > tau_lo0 can ever be non-zero ----
  float cand[CMAX];
#pragma unroll
  for (int k = 0; k < CMAX; ++k) cand[k] = -INFINITY;   // sentinel -> contributes 0
  int  cnt  = 0;
  bool over = false;
  for (int v = tid; v < DV; v += NT) {
    const float4 q = rv[v];
    const float xs0 = 0.5f * q.x, xs1 = 0.5f * q.y, xs2 = 0.5f * q.z, xs3 = 0.5f * q.w;
    const float xs[4] = {xs0, xs1, xs2, xs3};
#pragma unroll
    for (int j = 0; j < 4; ++j) {
      if (xs[j] > tau_lo0) {
        if (cnt < CMAX) cand[cnt] = xs[j]; else over = true;
        ++cnt;
      }
    }
  }
  if (over) atomicOr(&s_over, 1);
  __syncthreads();
  const bool full_scan = (s_over != 0);   // uniform across the block

  // ---- 50-step bisection ----
  float tau_lo = tau_lo0;
  const float tau_hi = max_s - TAU_HI_OFS;
  float dm    = tau_hi - tau_lo;
  float tau_m = tau_lo;
  float S     = 1.0f;

  for (int it = 0; it < N_ITER; ++it) {
    dm *= 0.5f;
    tau_m = tau_lo + dm;

    float part = 0.0f;
    if (full_scan) {
      for (int v = tid; v < DV; v += NT) {
        const float4 q = rv[v];
        float a;
        a = fmaxf(fmaf(0.5f, q.x, -tau_m), 0.0f); part = fmaf(a, a, part);
        a = fmaxf(fmaf(0.5f, q.y, -tau_m), 0.0f); part = fmaf(a, a, part);
        a = fmaxf(fmaf(0.5f, q.z, -tau_m), 0.0f); part = fmaf(a, a, part);
        a = fmaxf(fmaf(0.5f, q.w, -tau_m), 0.0f); part = fmaf(a, a, part);
      }
    } else {
#pragma unroll
      for (int k = 0; k < CMAX; ++k) {
        const float a = fmaxf(cand[k] - tau_m, 0.0f);   // sentinel -inf -> 0
        part = fmaf(a, a, part);
      }
    }

    // block sum, 2 barriers, double-banked scratch (bank = it&1)
    const int b = it & 1;
    part = wave_sum(part);
    if (lane == 0) red[b][wv] = part;
    __syncthreads();
    if (wv == 0) {
      float t = (lane < NWAVES) ? red[b][lane] : 0.0f;
      t = wave_sum(t);
      if (lane == 0) red[b][NWAVES] = t;
    }
    __syncthreads();
    const float sum = red[b][NWAVES];     // identical for all threads
    if (sum - 1.0f >= 0.0f) tau_lo = tau_m;
    S = sum;                              // sum of p for this iteration's tau_m
  }

  // ---- write normalized output: p = max(Xs - tau_m, 0)^2 / S ----
  const float invS = 1.0f / S;
  float4* __restrict__ ov = (float4*)dst;
  for (int v = tid; v < DV; v += NT) {
    const float4 q = rv[v];
    float4 o;
    float a;
    a = fmaxf(fmaf(0.5f, q.x, -tau_m), 0.0f); o.x = a * a * invS;
    a = fmaxf(fmaf(0.5f, q.y, -tau_m), 0.0f); o.y = a * a * invS;
    a = fmaxf(fmaf(0.5f, q.z, -tau_m), 0.0f); o.z = a * a * invS;
    a = fmaxf(fmaf(0.5f, q.w, -tau_m), 0.0f); o.w = a * a * invS;
    ov[v] = o;
  }
}

extern "C" void kernel_launch(void* const* d_in, const int* in_sizes, int n_in,
                              void* d_out, int out_size, void* d_ws, size_t ws_size,
                              hipStream_t stream) {
  (void)n_in; (void)d_ws; (void)ws_size; (void)out_size;
  const float* X = (const float*)d_in[0];
  float* Out     = (float*)d_out;
  const int rows = in_sizes[0] / D;   // 4096
  entmax15_bisect_kernel<<<rows, NT, 0, stream>>>(X, Out);
}